// PrecisionPredictorLoss_49847390437437
// MI455X (gfx1250) — compile-verified
//
#include <hip/hip_runtime.h>

typedef __attribute__((ext_vector_type(2))) float v2f;
typedef __attribute__((ext_vector_type(4))) float v4f;
typedef __attribute__((ext_vector_type(8))) float v8f;

#define NBLK 2048
#define TPB  256
#define ITERS 8               // TOTAL_F4 / (NBLK*TPB) exactly
#define TOTAL_F4 4194304      // 4096*64*64 / 4  == NBLK*TPB*ITERS
#define INV_N (1.0f / 16777216.0f)
#define GAMMA 0.1f
#define LAMBDA_2 0.1f

// Kernel 1: streaming sum((pred-true)^2) via V_WMMA_F32_16X16X4_F32.
// Passing the same v2f operand as A and B makes B == A^T in lane space,
// so D = A*A^T + C accumulates per-row sums of squares on the diagonal.
__global__ __launch_bounds__(TPB) void sumsq_wmma_kernel(
    const float* __restrict__ pred, const float* __restrict__ tru,
    float* __restrict__ partials) {
  v8f c = {};  // 16x16 f32 accumulator (diagonal carries the payload)
  const int tid = blockIdx.x * TPB + threadIdx.x;
  const v4f* p4 = reinterpret_cast<const v4f*>(pred);
  const v4f* t4 = reinterpret_cast<const v4f*>(tru);

  // Uniform, compile-time trip count: no per-iteration EXEC masking, and the
  // unroll lets the scheduler keep many b128 loads in flight (bandwidth-bound).
#pragma unroll
  for (int k = 0; k < ITERS; ++k) {
    const int i = tid + k * (NBLK * TPB);
    v4f a = __builtin_nontemporal_load(&p4[i]);  // global_load_b128, NT
    v4f b = __builtin_nontemporal_load(&t4[i]);  // global_load_b128, NT
    v2f d0 = { a[0] - b[0], a[1] - b[1] };
    v2f d1 = { a[2] - b[2], a[3] - b[3] };
    // 8-arg f32 pattern: (neg_a, A, neg_b, B, c_mod, C, reuse_a, reuse_b)
    c = __builtin_amdgcn_wmma_f32_16x16x4_f32(false, d0, false, d0, (short)0, c, false, false);
    c = __builtin_amdgcn_wmma_f32_16x16x4_f32(false, d1, false, d1, (short)0, c, false, false);
  }

  // Diagonal of 16x16 f32 C/D layout:
  //   r in [0,8):  D[r,r]     = VGPR r, lane r
  //   r in [0,8):  D[8+r,8+r] = VGPR r, lane 24+r
  const int lane = threadIdx.x & 31;
  float diag = 0.f;
#pragma unroll
  for (int r = 0; r < 8; ++r) {
    if (lane == r || lane == r + 24) diag += c[r];
  }
  // wave32 reduction
  for (int off = 16; off > 0; off >>= 1)
    diag += __shfl_down(diag, off, 32);

  __shared__ float wsum[TPB / 32];
  if (lane == 0) wsum[threadIdx.x >> 5] = diag;
  __syncthreads();
  if (threadIdx.x == 0) {
    float s = 0.f;
#pragma unroll
    for (int w = 0; w < TPB / 32; ++w) s += wsum[w];
    partials[blockIdx.x] = s;          // overwritten every launch: no ws zeroing needed
  }
}

// Kernel 2: deterministic final reduction + traces of first 32 matrices +
// 32x32 pairwise rank loss + output assembly.
__global__ __launch_bounds__(256) void finalize_kernel(
    const float* __restrict__ pred, const float* __restrict__ tru,
    const float* __restrict__ partials, float* __restrict__ out) {
  __shared__ float red[256];
  __shared__ float pl[32], tl[32], rowsum[32];
  const int t = threadIdx.x;

  // Fixed-order sum of the 2048 block partials (bit-deterministic).
  float s = 0.f;
  for (int i = t; i < NBLK; i += 256) s += partials[i];
  red[t] = s;
  __syncthreads();
  for (int h = 128; h > 0; h >>= 1) {
    if (t < h) red[t] += red[t + h];
    __syncthreads();
  }

  // Traces of the first MAX_RANK=32 matrices (64x64 each, diag stride 65).
  if (t < 32) {
    const float* pm = pred + (size_t)t * 4096;
    const float* tm = tru  + (size_t)t * 4096;
    float sp = 0.f, st = 0.f;
#pragma unroll 4
    for (int i = 0; i < 64; ++i) {
      sp += pm[i * 65];
      st += tm[i * 65];
    }
    pl[t] = sp;
    tl[t] = st;
  }
  __syncthreads();

  // Upper-triangle pairwise ranking hinge: row t handles pairs (t, j>t).
  if (t < 32) {
    const float pi = pl[t], ti = tl[t];
    float ci = 0.f;
    for (int j = t + 1; j < 32; ++j) {
      float dt = ti - tl[j];
      float dp = pi - pl[j];
      float v = 0.f;
      if (dt > 0.f)      v = fmaxf(-dp + GAMMA, 0.f);
      else if (dt < 0.f) v = fmaxf( dp + GAMMA, 0.f);
      ci += v;
    }
    rowsum[t] = ci;
  }
  __syncthreads();

  if (t == 0) {
    float rsum = 0.f;
#pragma unroll
    for (int i = 0; i < 32; ++i) rsum += rowsum[i];
    float rank_loss = rsum / 496.0f;        // m*(m-1)/2 = 496
    float eff       = red[0] * INV_N;
    float total     = eff + LAMBDA_2 * rank_loss;
    out[0] = total;
    out[1] = eff;
    out[2] = rank_loss;
  }
}

extern "C" void kernel_launch(void* const* d_in, const int* in_sizes, int n_in,
                              void* d_out, int out_size, void* d_ws, size_t ws_size,
                              hipStream_t stream) {
  const float* pred = (const float*)d_in[0];
  const float* tru  = (const float*)d_in[1];
  float* out        = (float*)d_out;
  float* partials   = (float*)d_ws;   // NBLK floats = 8 KB

  sumsq_wmma_kernel<<<NBLK, TPB, 0, stream>>>(pred, tru, partials);
  finalize_kernel<<<1, 256, 0, stream>>>(pred, tru, partials, out);
}